// BalanceMSELoss_85581518340365
// MI455X (gfx1250) — compile-verified
//
#include <hip/hip_runtime.h>

typedef float f4  __attribute__((ext_vector_type(4)));
typedef float v2f __attribute__((ext_vector_type(2)));
typedef float v8f __attribute__((ext_vector_type(8)));

#define BB   32
#define CC   3
#define HH   512
#define WW   512
#define W4   (WW / 4)                 // 128 float4 per row
#define NV4  (BB * CC * HH * W4)      // 6291456 float4 elements
#define NTOT ((double)(BB) * CC * HH * WW)  // 25165824

// ---------------------------------------------------------------- init ws
__global__ void bmse_init(double* __restrict__ ws) {
  if (threadIdx.x < 3) ws[threadIdx.x] = 0.0;
}

// ---------------------------------------------------------------- main reduction
// 2048 blocks x 256 threads (8 wave32) -> each thread handles exactly 12 float4s.
__global__ __launch_bounds__(256) void bmse_main(const float* __restrict__ inp,
                                                 const float* __restrict__ tgt,
                                                 const float* __restrict__ boxes,
                                                 double* __restrict__ ws) {
  __shared__ int4  sbox[BB];       // x1, x2, y1, y2 per batch
  __shared__ float red[3][64];     // 8 waves x 8 partials, per accumulator

  const int tid = threadIdx.x;

  // Precompute integer box bounds once per block (boxes is tiny, L2-resident).
  if (tid < BB) {
    float bx = boxes[tid * 4 + 0];
    float by = boxes[tid * 4 + 1];
    float bw = boxes[tid * 4 + 2];
    float bh = boxes[tid * 4 + 3];
    int x1 = (int)floorf(bx * (float)WW);
    int y1 = (int)floorf(by * (float)HH);
    int x2 = x1 + (int)floorf(bw * (float)WW);
    int y2 = y1 + (int)floorf(bh * (float)HH);
    sbox[tid] = make_int4(x1, x2, y1, y2);
  }
  __syncthreads();

  float st = 0.0f;  // sum of d^2 (total)
  float si = 0.0f;  // sum of d^2 inside box
  float ci = 0.0f;  // count inside box

  const int stride = gridDim.x * blockDim.x;
  const f4* __restrict__ in4 = (const f4*)inp;
  const f4* __restrict__ tg4 = (const f4*)tgt;

  for (int i = blockIdx.x * blockDim.x + tid; i < NV4; i += stride) {
    // i -> (b, y, x0): 65536 float4 per (b,c) plane; plane index bc = i >> 16
    int bc = i >> 16;
    int b  = bc / CC;                  // /3 -> mul+shift
    int y  = (i >> 7) & (HH - 1);
    int x0 = (i & (W4 - 1)) << 2;

    // Streaming, used-once data (201 MB > 192 MB L2) -> non-temporal loads.
    f4 a = __builtin_nontemporal_load(in4 + i);
    f4 t = __builtin_nontemporal_load(tg4 + i);

    int4 bx   = sbox[b];
    bool yin  = (y >= bx.z) & (y <= bx.w);

    f4 d  = a - t;
    f4 d2 = d * d;

#pragma unroll
    for (int j = 0; j < 4; ++j) {
      int  x  = x0 + j;
      bool in = yin & (x >= bx.x) & (x <= bx.y);
      st += d2[j];
      si += in ? d2[j] : 0.0f;
      ci += in ? 1.0f  : 0.0f;
    }
  }

  // ---- wave-level partial reduce: 32 lanes -> 8 distinct group sums (xor 16, xor 8)
  st += __shfl_xor(st, 16); st += __shfl_xor(st, 8);
  si += __shfl_xor(si, 16); si += __shfl_xor(si, 8);
  ci += __shfl_xor(ci, 16); ci += __shfl_xor(ci, 8);

  const int w = tid >> 5;
  const int l = tid & 31;
  if (l < 8) {
    red[0][w * 8 + l] = st;
    red[1][w * 8 + l] = si;
    red[2][w * 8 + l] = ci;
  }
  __syncthreads();

  // ---- block-level reduce via V_WMMA_F32_16X16X4_F32.
  // A = ones(16x4)  =>  D[m][n] = sum_k B[k][n].  Summing row 0 over its 16
  // columns yields the sum of ALL 64 B slots, independent of B's lane mapping.
  // All 8 waves execute this uniformly so EXEC is all-1s (WMMA requirement).
  v2f ones; ones.x = 1.0f; ones.y = 1.0f;
  float tot[3];
#pragma unroll
  for (int acc = 0; acc < 3; ++acc) {
    v2f bmat;
    bmat.x = red[acc][l];
    bmat.y = red[acc][32 + l];
    v8f cz = {0.f, 0.f, 0.f, 0.f, 0.f, 0.f, 0.f, 0.f};
    v8f dm = __builtin_amdgcn_wmma_f32_16x16x4_f32(
        /*neg_a=*/false, ones, /*neg_b=*/false, bmat,
        /*c_mod=*/(short)0, cz, /*reuse_a=*/false, /*reuse_b=*/false);
    // D VGPR0: lanes 0-15 hold row M=0, cols N=0..15 (all rows equal colsums).
    float v = dm[0];
    v += __shfl_xor(v, 1);
    v += __shfl_xor(v, 2);
    v += __shfl_xor(v, 4);
    v += __shfl_xor(v, 8);   // lane 0 = sum over N=0..15 = block total
    tot[acc] = v;
  }

  if (tid == 0) {
    // fp64 accumulation across 2048 blocks -> GLOBAL_ATOMIC_ADD_F64
    __hip_atomic_fetch_add(&ws[0], (double)tot[0], __ATOMIC_RELAXED, __HIP_MEMORY_SCOPE_AGENT);
    __hip_atomic_fetch_add(&ws[1], (double)tot[1], __ATOMIC_RELAXED, __HIP_MEMORY_SCOPE_AGENT);
    __hip_atomic_fetch_add(&ws[2], (double)tot[2], __ATOMIC_RELAXED, __HIP_MEMORY_SCOPE_AGENT);
  }
}

// ---------------------------------------------------------------- finalize
__global__ void bmse_final(const double* __restrict__ ws, float* __restrict__ out) {
  double st = ws[0];
  double si = ws[1];
  double ci = ws[2];
  double inside_loss  = si / ci;
  double outside_loss = (st - si) / (NTOT - ci);
  out[0] = (float)(0.5 * inside_loss + 0.5 * outside_loss);  // ETA = 1.0
}

// ---------------------------------------------------------------- launch
extern "C" void kernel_launch(void* const* d_in, const int* in_sizes, int n_in,
                              void* d_out, int out_size, void* d_ws, size_t ws_size,
                              hipStream_t stream) {
  (void)in_sizes; (void)n_in; (void)out_size; (void)ws_size;
  const float* inp   = (const float*)d_in[0];
  const float* tgt   = (const float*)d_in[1];
  const float* boxes = (const float*)d_in[2];
  float*  out = (float*)d_out;
  double* ws  = (double*)d_ws;

  bmse_init<<<1, 64, 0, stream>>>(ws);
  bmse_main<<<2048, 256, 0, stream>>>(inp, tgt, boxes, ws);
  bmse_final<<<1, 1, 0, stream>>>(ws, out);
}